// Mono3DRoIHeads_81449759801385
// MI455X (gfx1250) — compile-verified
//
#include <hip/hip_runtime.h>
#include <hip/hip_bf16.h>
#include <math.h>

#define NBOX 256
#define HIMG 512
#define WIMG 512
#define KIN  1037
#define KPAD 1040
#define DHID 512
#define BIGF 3.4e+38f

typedef float v2f __attribute__((ext_vector_type(2)));
typedef float v8f __attribute__((ext_vector_type(8)));

__device__ __forceinline__ float softplus_f(float x) {
    return (x > 20.0f) ? x : log1pf(expf(x));
}

// ---------------------------------------------------------------------------
// Kernel 0: zero-pad W_ctx [1037,512] -> Wp [1040,512]
// ---------------------------------------------------------------------------
__global__ __launch_bounds__(256) void pack_w_kernel(const float* __restrict__ Wsrc,
                                                     float* __restrict__ Wp) {
    int i = blockIdx.x * 256 + threadIdx.x;      // 0 .. 1040*512-1 (exact grid)
    int row = i / DHID;
    Wp[i] = (row < KIN) ? Wsrc[i] : 0.0f;
}

// ---------------------------------------------------------------------------
// Kernel 1: per-box depth stats (median via radix-select, mean, std, min, max)
//           + pack x_in row = [shared_features | bbox | intrinsics | stats | 0pad]
// One workgroup (256 threads) per box.
// ---------------------------------------------------------------------------
__global__ __launch_bounds__(256) void stats_pack_kernel(
    const float* __restrict__ sf,      // [256,1024]
    const float* __restrict__ bbox,    // [256,4]
    const float* __restrict__ intr,    // [256,4]
    const float* __restrict__ depth,   // [512,512]
    float* __restrict__ xin,           // [256,1040]
    float* __restrict__ stats_ws,      // [256,5]
    float* __restrict__ out)           // [256,30]
{
    const int n = blockIdx.x;
    const int tid = threadIdx.x;

    __shared__ float sred[256];
    __shared__ int   ired[256];
    __shared__ unsigned s_prefix;
    __shared__ int      s_k;

    const float bx1 = bbox[n*4+0], by1 = bbox[n*4+1];
    const float bx2 = bbox[n*4+2], by2 = bbox[n*4+3];
    const int ix1 = (int)bx1, iy1 = (int)by1, ix2 = (int)bx2, iy2 = (int)by2;
    const int x1 = min(max(ix1, 0), WIMG - 1);
    const int x2 = max(x1 + 1, min(ix2, WIMG));
    const int y1 = min(max(iy1, 0), HIMG - 1);
    const int y2 = max(y1 + 1, min(iy2, HIMG));
    const int bw = x2 - x1;
    const int area = bw * (y2 - y1);

    // ---- pass 1: sum / min / max -----------------------------------------
    float lsum = 0.0f, lmin = BIGF, lmax = -BIGF;
    for (int i = tid; i < area; i += 256) {
        int yy = y1 + i / bw, xx = x1 + i % bw;
        float d = depth[yy * WIMG + xx];
        lsum += d;
        lmin = fminf(lmin, d);
        lmax = fmaxf(lmax, d);
    }
    sred[tid] = lsum; __syncthreads();
    for (int s = 128; s > 0; s >>= 1) { if (tid < s) sred[tid] += sred[tid + s]; __syncthreads(); }
    const float sum = sred[0]; __syncthreads();
    sred[tid] = lmin; __syncthreads();
    for (int s = 128; s > 0; s >>= 1) { if (tid < s) sred[tid] = fminf(sred[tid], sred[tid + s]); __syncthreads(); }
    const float vmin = sred[0]; __syncthreads();
    sred[tid] = lmax; __syncthreads();
    for (int s = 128; s > 0; s >>= 1) { if (tid < s) sred[tid] = fmaxf(sred[tid], sred[tid + s]); __syncthreads(); }
    const float vmax = sred[0]; __syncthreads();

    const float mean = sum / (float)area;

    // ---- pass 2: two-pass variance (unbiased, matches reference) ---------
    float lss = 0.0f;
    for (int i = tid; i < area; i += 256) {
        int yy = y1 + i / bw, xx = x1 + i % bw;
        float t = depth[yy * WIMG + xx] - mean;
        lss += t * t;
    }
    sred[tid] = lss; __syncthreads();
    for (int s = 128; s > 0; s >>= 1) { if (tid < s) sred[tid] += sred[tid + s]; __syncthreads(); }
    const float var = sred[0] / fmaxf((float)area - 1.0f, 1.0f);
    const float stdv = (area > 1) ? sqrtf(var) : 0.0f;
    __syncthreads();

    // ---- exact lower median via bitwise radix-select ---------------------
    // depth > 0 always, so float bits are monotone as uint32 (bit31 == 0).
    if (tid == 0) { s_prefix = 0u; s_k = (area - 1) >> 1; }
    __syncthreads();
    for (int b = 30; b >= 0; --b) {
        const unsigned hi = s_prefix >> (b + 1);
        int lc = 0;
        for (int i = tid; i < area; i += 256) {
            int yy = y1 + i / bw, xx = x1 + i % bw;
            unsigned u = __float_as_uint(depth[yy * WIMG + xx]);
            if ((u >> (b + 1)) == hi && ((u >> b) & 1u) == 0u) lc++;
        }
        ired[tid] = lc; __syncthreads();
        for (int s = 128; s > 0; s >>= 1) { if (tid < s) ired[tid] += ired[tid + s]; __syncthreads(); }
        if (tid == 0) {
            int c = ired[0];
            if (s_k >= c) { s_k -= c; s_prefix |= (1u << b); }
        }
        __syncthreads();
    }
    const float med = __uint_as_float(s_prefix);

    const float st[5] = { med, mean, stdv, vmin, vmax };
    if (tid == 0) {
        #pragma unroll
        for (int j = 0; j < 5; ++j) {
            stats_ws[n * 5 + j] = st[j];
            out[n * 30 + 25 + j] = st[j];
        }
    }

    // ---- pack x_in row ---------------------------------------------------
    for (int i = tid; i < 1024; i += 256) xin[n * KPAD + i] = sf[n * 1024 + i];
    if (tid < 16) {
        float v;
        if (tid < 4)       v = bbox[n * 4 + tid];
        else if (tid < 8)  v = intr[n * 4 + (tid - 4)];
        else if (tid < 13) v = st[tid - 8];
        else               v = 0.0f;                 // K padding 1037..1039
        xin[n * KPAD + 1024 + tid] = v;
    }
}

// ---------------------------------------------------------------------------
// Kernel 2: hpre = x_in @ Wp + b_ctx via V_WMMA_F32_16X16X4_F32
// 512 output tiles of 16x16; one wave per tile; 64 blocks x 8 waves.
// ---------------------------------------------------------------------------
__global__ __launch_bounds__(256) void gemm_wmma_kernel(
    const float* __restrict__ xin,    // [256,1040]
    const float* __restrict__ Wp,     // [1040,512]
    const float* __restrict__ bctx,   // [512]
    float* __restrict__ hpre)         // [256,512]
{
    const int wave = (blockIdx.x << 3) | (threadIdx.x >> 5);   // 0..511
    const int lane = threadIdx.x & 31;
    const int half = lane >> 4;        // 0: K pair {0,1}, 1: K pair {2,3}
    const int l16  = lane & 15;
    const int tm = wave >> 5;          // 0..15  (row tile)
    const int tn = wave & 31;          // 0..31  (col tile)

    const float* arow = xin + (tm * 16 + l16) * KPAD;   // A row for this lane
    const float* bcol = Wp + (tn * 16 + l16);           // B column for this lane

    v8f acc = {};
    for (int k = 0; k < KPAD; k += 4) {
        const int ka = k + half * 2;
        v2f a, b;
        a.x = arow[ka];
        a.y = arow[ka + 1];
        b.x = bcol[ka * DHID];
        b.y = bcol[(ka + 1) * DHID];
        // D = A(16x4,f32) * B(4x16,f32) + C  -> v_wmma_f32_16x16x4_f32
        acc = __builtin_amdgcn_wmma_f32_16x16x4_f32(
            false, a, false, b, (short)0, acc, false, false);
    }

    const int col = tn * 16 + l16;
    const float bias = bctx[col];
    #pragma unroll
    for (int r = 0; r < 8; ++r) {
        const int row = tm * 16 + half * 8 + r;   // C/D layout: lanes 16-31 hold M+8
        hpre[row * DHID + col] = acc[r] + bias;
    }
}

// ---------------------------------------------------------------------------
// Kernel 3: LayerNorm (biased var) + affine + ReLU, one block per row
// ---------------------------------------------------------------------------
__global__ __launch_bounds__(256) void ln_relu_kernel(
    const float* __restrict__ hpre, const float* __restrict__ g,
    const float* __restrict__ bln, float* __restrict__ xact)
{
    const int n = blockIdx.x, tid = threadIdx.x;
    __shared__ float red[256];

    const float h0 = hpre[n * DHID + tid];
    const float h1 = hpre[n * DHID + 256 + tid];

    red[tid] = h0 + h1; __syncthreads();
    for (int s = 128; s > 0; s >>= 1) { if (tid < s) red[tid] += red[tid + s]; __syncthreads(); }
    const float mean = red[0] * (1.0f / 512.0f); __syncthreads();

    const float d0 = h0 - mean, d1 = h1 - mean;
    red[tid] = d0 * d0 + d1 * d1; __syncthreads();
    for (int s = 128; s > 0; s >>= 1) { if (tid < s) red[tid] += red[tid + s]; __syncthreads(); }
    const float var = red[0] * (1.0f / 512.0f);
    const float rs = rsqrtf(var + 1e-5f);

    xact[n * DHID + tid]       = fmaxf(d0 * rs * g[tid] + bln[tid], 0.0f);
    xact[n * DHID + 256 + tid] = fmaxf(d1 * rs * g[256 + tid] + bln[256 + tid], 0.0f);
}

// ---------------------------------------------------------------------------
// Kernel 4: 9 head dot-products (512-dim) + box geometry, one block per row
// ---------------------------------------------------------------------------
__global__ __launch_bounds__(64) void heads_kernel(
    const float* __restrict__ xact,
    const float* __restrict__ Wdim, const float* __restrict__ bdim,
    const float* __restrict__ Wrot, const float* __restrict__ brot,
    const float* __restrict__ Wdep, const float* __restrict__ bdep,
    const float* __restrict__ Woff, const float* __restrict__ boff,
    const float* __restrict__ Wmu,  const float* __restrict__ bmu,
    const float* __restrict__ bbox, const float* __restrict__ intr,
    float* __restrict__ out)
{
    const int n = blockIdx.x, tid = threadIdx.x;
    __shared__ float red[9][64];

    float acc[9];
    #pragma unroll
    for (int j = 0; j < 9; ++j) acc[j] = 0.0f;

    for (int k = tid; k < DHID; k += 64) {
        const float xv = xact[n * DHID + k];
        acc[0] += xv * Wdim[k * 3 + 0];
        acc[1] += xv * Wdim[k * 3 + 1];
        acc[2] += xv * Wdim[k * 3 + 2];
        acc[3] += xv * Wrot[k * 2 + 0];
        acc[4] += xv * Wrot[k * 2 + 1];
        acc[5] += xv * Wdep[k];
        acc[6] += xv * Woff[k * 2 + 0];
        acc[7] += xv * Woff[k * 2 + 1];
        acc[8] += xv * Wmu[k];
    }
    #pragma unroll
    for (int j = 0; j < 9; ++j) red[j][tid] = acc[j];
    __syncthreads();
    if (tid < 9) {
        float s = 0.0f;
        for (int t = 0; t < 64; ++t) s += red[tid][t];
        red[tid][0] = s;
    }
    __syncthreads();

    if (tid == 0) {
        float dims[3];
        #pragma unroll
        for (int j = 0; j < 3; ++j) dims[j] = softplus_f(red[j][0] + bdim[j]);
        const float r0 = red[3][0] + brot[0];
        const float r1 = red[4][0] + brot[1];
        const float nrm = fmaxf(sqrtf(r0 * r0 + r1 * r1), 1e-12f);
        const float sinv = r0 / nrm, cosv = r1 / nrm;
        const float depthv = softplus_f(red[5][0] + bdep[0]) + 1e-4f;
        const float off0 = red[6][0] + boff[0];
        const float off1 = red[7][0] + boff[1];
        const float mu = red[8][0] + bmu[0];

        const float fx = intr[n*4+0], fy = intr[n*4+1];
        const float px = intr[n*4+2], py = intr[n*4+3];
        const float u = (bbox[n*4+0] + bbox[n*4+2]) * 0.5f + off0;
        const float v = (bbox[n*4+1] + bbox[n*4+3]) * 0.5f + off1;
        const float x_c = (u - px) * depthv / fx;
        const float y_c = (v - py) * depthv / fy;

        const float sx[8] = { 1, 1,-1,-1, 1, 1,-1,-1};
        const float sy[8] = { 1,-1,-1, 1, 1,-1,-1, 1};
        const float sz[8] = { 1, 1, 1, 1,-1,-1,-1,-1};
        #pragma unroll
        for (int i = 0; i < 8; ++i) {
            const float xc = dims[0] * 0.5f * sx[i];
            const float yc = dims[1] * 0.5f * sy[i];
            const float zc = dims[2] * 0.5f * sz[i];
            const float xr = xc * cosv - yc * sinv;
            const float yr = xc * sinv + yc * cosv;
            out[n*30 + 3*i + 0] = xr + x_c;
            out[n*30 + 3*i + 1] = yr + y_c;
            out[n*30 + 3*i + 2] = zc + depthv;
        }
        out[n*30 + 24] = mu;
    }
}

// ---------------------------------------------------------------------------
extern "C" void kernel_launch(void* const* d_in, const int* in_sizes, int n_in,
                              void* d_out, int out_size, void* d_ws, size_t ws_size,
                              hipStream_t stream) {
    const float* sf    = (const float*)d_in[0];   // shared_features [256,1024]
    const float* bbox  = (const float*)d_in[1];   // bbox_2d [256,4]
    const float* intr  = (const float*)d_in[2];   // camera_intrinsics [256,4]
    const float* depth = (const float*)d_in[3];   // depth_map [512,512]
    const float* Wctx  = (const float*)d_in[4];   // [1037,512]
    const float* bctx  = (const float*)d_in[5];   // [512]
    const float* ln_g  = (const float*)d_in[6];
    const float* ln_b  = (const float*)d_in[7];
    const float* Wdim  = (const float*)d_in[8];
    const float* bdim  = (const float*)d_in[9];
    const float* Wrot  = (const float*)d_in[10];
    const float* brot  = (const float*)d_in[11];
    const float* Wdep  = (const float*)d_in[12];
    const float* bdep  = (const float*)d_in[13];
    const float* Woff  = (const float*)d_in[14];
    const float* boff  = (const float*)d_in[15];
    const float* Wmu   = (const float*)d_in[16];
    const float* bmu   = (const float*)d_in[17];
    float* out = (float*)d_out;

    // workspace layout (floats)
    float* ws    = (float*)d_ws;
    float* xin   = ws;                                  // 256*1040
    float* Wp    = xin + NBOX * KPAD;                   // 1040*512
    float* hpre  = Wp + KPAD * DHID;                    // 256*512
    float* xact  = hpre + NBOX * DHID;                  // 256*512
    float* stats = xact + NBOX * DHID;                  // 256*5

    pack_w_kernel<<<(KPAD * DHID) / 256, 256, 0, stream>>>(Wctx, Wp);
    stats_pack_kernel<<<NBOX, 256, 0, stream>>>(sf, bbox, intr, depth, xin, stats, out);
    gemm_wmma_kernel<<<64, 256, 0, stream>>>(xin, Wp, bctx, hpre);
    ln_relu_kernel<<<NBOX, 256, 0, stream>>>(hpre, ln_g, ln_b, xact);
    heads_kernel<<<NBOX, 64, 0, stream>>>(xact, Wdim, bdim, Wrot, brot, Wdep, bdep,
                                          Woff, boff, Wmu, bmu, bbox, intr, out);
}